// CRF_1752346656948
// MI455X (gfx1250) — compile-verified
//
#include <hip/hip_runtime.h>
#include <hip/hip_bf16.h>

typedef __attribute__((ext_vector_type(2))) float v2f;
typedef __attribute__((ext_vector_type(8))) float v8f;

#define B_TOT 512
#define T_LEN 512
#define L_DIM 64
#define TILE_B 16

// CRF forward (normalizer) via exp-space GEMM + gold-path score.
// One block = 16 batches. 4 waves (wave32); wave w owns output states [16w,16w+16).
__global__ __launch_bounds__(128) void crf_fwd_kernel(
    const float* __restrict__ emission,    // [B,T,L]
    const int*   __restrict__ target,      // [B,T]
    const float* __restrict__ mask,        // [B,T]
    const float* __restrict__ start_trans, // [L]
    const float* __restrict__ trans,       // [L,L]
    const float* __restrict__ end_trans,   // [L]
    float* __restrict__ out_per_batch)     // [B] : normalizer - path
{
    __shared__ float sS[TILE_B][L_DIM];   // running scores
    __shared__ float sPt[L_DIM][TILE_B];  // exp(s - m), transposed (k-major) for A frags
    __shared__ float sM[TILE_B];          // per-row max / later per-row logsumexp
    __shared__ float sRed[TILE_B][8];     // reduction scratch

    const int tid  = threadIdx.x;
    const int wave = tid >> 5;
    const int lane = tid & 31;
    const int lidx = lane & 15;
    const int h    = lane >> 4;           // lane half (ISA A/B/C layout)
    const int b0   = blockIdx.x * TILE_B;
    const int col  = wave * 16 + lidx;    // this lane's output state j

    // Preload B fragments: E = exp(trans), B is 4x16 per WMMA chunk.
    // B layout (4x16 f32): VGPR0 = K = 0 (lanes 0-15) / K = 2 (lanes 16-31); VGPR1 = K = 1 / K = 3.
    v2f bfrag[16];
    #pragma unroll
    for (int kk = 0; kk < 16; ++kk) {
        int k0 = 4 * kk + 2 * h;
        bfrag[kk].x = __expf(trans[k0 * L_DIM + col]);
        bfrag[kk].y = __expf(trans[(k0 + 1) * L_DIM + col]);
    }
    const float et = end_trans[col];

    // init: s = start_trans + emission[:, 0, :]
    for (int idx = tid; idx < TILE_B * L_DIM; idx += 128) {
        int r = idx >> 6, j = idx & 63;
        sS[r][j] = start_trans[j] + emission[(size_t)(b0 + r) * T_LEN * L_DIM + j];
    }
    __syncthreads();

    for (int t = 1; t < T_LEN; ++t) {
        // ---- 1. per-row max (8 threads per row, then reduce) ----
        {
            int r = tid >> 3, seg = tid & 7;
            float mx = -3.4e38f;
            #pragma unroll
            for (int q = 0; q < 8; ++q) mx = fmaxf(mx, sS[r][seg * 8 + q]);
            sRed[r][seg] = mx;
        }
        __syncthreads();
        if (tid < TILE_B) {
            float mx = sRed[tid][0];
            #pragma unroll
            for (int q = 1; q < 8; ++q) mx = fmaxf(mx, sRed[tid][q]);
            sM[tid] = mx;
        }
        __syncthreads();

        // ---- 2. P^T = exp(s - m), stored k-major so A-frag reads are conflict-light ----
        for (int idx = tid; idx < TILE_B * L_DIM; idx += 128) {
            int r = idx >> 6, j = idx & 63;
            sPt[j][r] = __expf(sS[r][j] - sM[r]);
        }
        __syncthreads();

        // ---- 3. D(16x16) = P(16x64) @ E(64x16-tile): 16 chained V_WMMA_F32_16X16X4_F32 ----
        v8f c = {0.f, 0.f, 0.f, 0.f, 0.f, 0.f, 0.f, 0.f};
        #pragma unroll
        for (int kk = 0; kk < 16; ++kk) {
            int k0 = 4 * kk + 2 * h;
            v2f a;                         // A 16x4: VGPR0 = K even-half, VGPR1 = K+1
            a.x = sPt[k0][lidx];
            a.y = sPt[k0 + 1][lidx];
            c = __builtin_amdgcn_wmma_f32_16x16x4_f32(
                    false, a, false, bfrag[kk], (short)0, c, false, false);
        }

        // ---- 4. s' = mask*(m + log(D) + em) + (1-mask)*s + end_mask*end_trans ----
        // D layout: lane holds rows v + 8h (v = VGPR index), column = lidx.
        #pragma unroll
        for (int v = 0; v < 8; ++v) {
            int r = v + 8 * h;
            size_t bt = (size_t)(b0 + r) * T_LEN + t;
            if (t + 1 < T_LEN)
                __builtin_prefetch(&emission[(bt + 1) * L_DIM + col], 0, 1);
            float mv  = mask[bt];
            float mn  = (t + 1 < T_LEN) ? mask[bt + 1] : 0.0f;
            float e   = (mv > mn) ? 1.0f : 0.0f;
            float em  = emission[bt * L_DIM + col];
            float nxt = sM[r] + __logf(c[v]) + em;
            float old = sS[r][col];
            sS[r][col] = mv * nxt + (1.0f - mv) * old + e * et;
        }
        __syncthreads();
    }

    // ---- normalizer = logsumexp_j s[b, j] ----
    if (tid < TILE_B) {
        float mx = -3.4e38f;
        for (int j = 0; j < L_DIM; ++j) mx = fmaxf(mx, sS[tid][j]);
        float s = 0.f;
        for (int j = 0; j < L_DIM; ++j) s += __expf(sS[tid][j] - mx);
        sM[tid] = mx + __logf(s);
    }
    __syncthreads();

    // ---- gold path score: separable sum over t (prev = tgt[t-1] unconditionally) ----
    {
        int r = tid >> 3, seg = tid & 7;
        size_t brow = (size_t)(b0 + r) * T_LEN;
        float acc = 0.f;
        for (int t = 1 + seg; t < T_LEN; t += 8) {
            int prev = target[brow + t - 1];
            int cur  = target[brow + t];
            float mv = mask[brow + t];
            float mn = (t + 1 < T_LEN) ? mask[brow + t + 1] : 0.0f;
            float e  = (mv > mn) ? 1.0f : 0.0f;
            acc += mv * (trans[prev * L_DIM + cur] +
                         emission[(brow + t) * L_DIM + cur])
                 + e * end_trans[cur];
        }
        sRed[r][seg] = acc;
    }
    __syncthreads();
    if (tid < TILE_B) {
        size_t brow = (size_t)(b0 + tid) * T_LEN;
        int t0 = target[brow];
        float ps = start_trans[t0] + emission[brow * L_DIM + t0];
        #pragma unroll
        for (int q = 0; q < 8; ++q) ps += sRed[tid][q];
        out_per_batch[b0 + tid] = sM[tid] - ps;
    }
}

// mean over 512 per-batch values -> scalar
__global__ __launch_bounds__(256) void crf_mean_kernel(const float* __restrict__ in,
                                                       float* __restrict__ out) {
    __shared__ float red[256];
    int tid = threadIdx.x;
    red[tid] = in[tid] + in[tid + 256];
    __syncthreads();
    for (int s = 128; s > 0; s >>= 1) {
        if (tid < s) red[tid] += red[tid + s];
        __syncthreads();
    }
    if (tid == 0) out[0] = red[0] * (1.0f / 512.0f);
}

extern "C" void kernel_launch(void* const* d_in, const int* in_sizes, int n_in,
                              void* d_out, int out_size, void* d_ws, size_t ws_size,
                              hipStream_t stream) {
    const float* emission    = (const float*)d_in[0];
    const int*   target      = (const int*)d_in[1];
    const float* mask        = (const float*)d_in[2];
    const float* start_trans = (const float*)d_in[3];
    const float* trans       = (const float*)d_in[4];
    const float* end_trans   = (const float*)d_in[5];
    float* per_batch = (float*)d_ws;       // 512 floats of scratch
    float* out = (float*)d_out;

    crf_fwd_kernel<<<B_TOT / TILE_B, 128, 0, stream>>>(
        emission, target, mask, start_trans, trans, end_trans, per_batch);
    crf_mean_kernel<<<1, 256, 0, stream>>>(per_batch, out);
}